// BimodalCompressor_5385888989206
// MI455X (gfx1250) — compile-verified
//
#include <hip/hip_runtime.h>
#include <cmath>

typedef float v2f __attribute__((ext_vector_type(2)));
typedef float v8f __attribute__((ext_vector_type(8)));

// D = A(16x4) * B(4x16) + C, fp32, wave32.
#define WMMA_F32(a, b, c) \
    __builtin_amdgcn_wmma_f32_16x16x4_f32(false, (a), false, (b), (short)0, (c), false, false)

constexpr int Npts = 40000;   // N
constexpr int KNb  = 32;      // neighbors per point
constexpr int Pk   = 15;      // kernel points
constexpr int Cch  = 128;     // channels
constexpr float SLOPE = 0.01f;
constexpr float BNEPS = 1e-5f;

// ---------------------------------------------------------------------------
// fp32 WMMA GEMM:  C[M,Nc] (+)= A[M,K] @ B[K,Nc]   (row-major)
// Block = 4 waves = 64(M) x 64(N).  Each wave: 4 stacked 16x16 M-subtiles on a
// single 16-col strip -> B fragment loaded once per k-step, reused 4x.
// Main loop is guard-free (K & ~3); guarded tail only for K=3 / K=9 layers.
// ---------------------------------------------------------------------------
__global__ __launch_bounds__(128)
void gemm_f32_wmma(const float* __restrict__ A, int lda,
                   const float* __restrict__ Bp, int ldb,
                   float* __restrict__ C, int ldc,
                   int M, int K, int Nc, int accumulate)
{
    const int wave = threadIdx.x >> 5;
    const int lane = threadIdx.x & 31;
    const int half = lane >> 4;   // 0: lanes 0-15, 1: lanes 16-31
    const int l16  = lane & 15;
    const int m0 = blockIdx.x * 64;
    const int n0 = blockIdx.y * 64 + wave * 16;
    if (n0 >= Nc) return;                      // wave-uniform

    const int bcol = n0 + l16;
    v8f acc[4] = {{}, {}, {}, {}};

    auto step = [&](int kk) {                  // one unconditional k-step of 4
        const int ka = kk + 2 * half;
        v2f b;
        b.x = Bp[(size_t)ka * ldb + bcol];
        b.y = Bp[(size_t)(ka + 1) * ldb + bcol];
#pragma unroll
        for (int mi = 0; mi < 4; ++mi) {
            const int row = m0 + mi * 16 + l16;
            v2f a;
            a.x = A[(size_t)row * lda + ka];
            a.y = A[(size_t)row * lda + ka + 1];
            acc[mi] = WMMA_F32(a, b, acc[mi]);
        }
    };

    const int Kmain = K & ~3;
    int k = 0;
    for (; k + 8 <= Kmain; k += 8) { step(k); step(k + 4); }   // 2x unrolled
    for (; k < Kmain; k += 4)      { step(k); }
    if (k < K) {                                               // guarded tail
        const int ka = k + 2 * half;
        v2f a, b;
        a.x = b.x = 0.0f; a.y = b.y = 0.0f;
        if (ka < K)     { b.x = Bp[(size_t)ka * ldb + bcol]; }
        if (ka + 1 < K) { b.y = Bp[(size_t)(ka + 1) * ldb + bcol]; }
#pragma unroll
        for (int mi = 0; mi < 4; ++mi) {
            const int row = m0 + mi * 16 + l16;
            v2f a2 = a;
            if (ka < K)     a2.x = A[(size_t)row * lda + ka];
            if (ka + 1 < K) a2.y = A[(size_t)row * lda + ka + 1];
            acc[mi] = WMMA_F32(a2, b, acc[mi]);
        }
    }

#pragma unroll
    for (int mi = 0; mi < 4; ++mi) {
#pragma unroll
        for (int r = 0; r < 8; ++r) {
            const int row = m0 + mi * 16 + r + 8 * half;  // D: VGPR r -> rows r / r+8
            float v = acc[mi][r];
            if (accumulate) v += C[(size_t)row * ldc + bcol];
            C[(size_t)row * ldc + bcol] = v;
        }
    }
}

// ---------------------------------------------------------------------------
// Batch statistics, two deterministic phases (fully coalesced reads).
// Phase 1: 256 blocks grid-stride the flat [M,Cc] array.  Since 256 and the
// grid stride are multiples of Cc (Cc in {64,128,256}), each thread owns one
// fixed column.  Block-level LDS tree -> part[block][2][Cc].
// ---------------------------------------------------------------------------
__global__ __launch_bounds__(256)
void colstats_part(const float* __restrict__ X, int M, int Cc,
                   float* __restrict__ part)
{
    const int tid = threadIdx.x;
    float s = 0.0f, s2 = 0.0f;
    const int total = M * Cc;
    for (int i = blockIdx.x * 256 + tid; i < total; i += 256 * 256) {
        const float v = X[i];
        s += v; s2 += v * v;
    }
    __shared__ float rs[256], rq[256];
    rs[tid] = s; rq[tid] = s2;
    __syncthreads();
    if (tid < Cc) {
        float ts = 0.0f, tq = 0.0f;
        for (int t = tid; t < 256; t += Cc) { ts += rs[t]; tq += rq[t]; }
        part[(size_t)blockIdx.x * 2 * Cc + tid]      = ts;
        part[(size_t)blockIdx.x * 2 * Cc + Cc + tid] = tq;
    }
}

// Phase 2: fold 256 partials per column in fixed order -> mean / biased var.
__global__ __launch_bounds__(64)
void colstats_final(const float* __restrict__ part, int Cc, int M,
                    float* __restrict__ stats)
{
    const int c = blockIdx.x * 64 + threadIdx.x;
    if (c >= Cc) return;
    float s = 0.0f, q = 0.0f;
    for (int b = 0; b < 256; ++b) {
        s += part[(size_t)b * 2 * Cc + c];
        q += part[(size_t)b * 2 * Cc + Cc + c];
    }
    const float m = s / (float)M;
    stats[c] = m;
    stats[Cc + c] = q / (float)M - m * m;
}

// ---------------------------------------------------------------------------
// BN (training-mode batch stats) + LeakyReLU, elementwise
// ---------------------------------------------------------------------------
__global__ __launch_bounds__(256)
void bn_lrelu(const float* __restrict__ raw, float* __restrict__ out,
              const float* __restrict__ g, const float* __restrict__ b,
              const float* __restrict__ stats, int total, int Cc)
{
    int i = blockIdx.x * 256 + threadIdx.x;
    if (i >= total) return;
    const int c = i % Cc;
    float x = raw[i];
    x = (x - stats[c]) * rsqrtf(stats[Cc + c] + BNEPS) * g[c] + b[c];
    out[i] = (x >= 0.0f) ? x : SLOPE * x;
}

// ---------------------------------------------------------------------------
// Row sums of the [N,128] feature matrix (density-normalization count input)
// ---------------------------------------------------------------------------
__global__ __launch_bounds__(128)
void rowsum_kernel(const float* __restrict__ X, float* __restrict__ out)
{
    const int row  = blockIdx.x * 4 + (threadIdx.x >> 5);
    const int lane = threadIdx.x & 31;
    const float* p = X + (size_t)row * Cch;
    float s = p[lane] + p[lane + 32] + p[lane + 64] + p[lane + 96];
    for (int off = 16; off; off >>= 1) s += __shfl_down(s, off, 32);
    if (lane == 0) out[row] = s;
}

// ---------------------------------------------------------------------------
// Fused rigid KPConv.  Block = 16 points, 128 threads (4 waves).
// Stage 1 (per point): weighted[P,C] = infl^T[16,32] @ nx[32,128]  (WMMA)
// Stage 2 (per tile):  conv[16,C]   += weighted[:,p,:] @ W[p]      (WMMA)
// A fragments shared across each wave's two column tiles (2 WMMAs / fragment).
// LDS strides padded to kill 64-bank conflicts.
// ---------------------------------------------------------------------------
__global__ __launch_bounds__(128)
void kpconv_kernel(const float* __restrict__ feat,
                   const float* __restrict__ pts,
                   const int*   __restrict__ nbr,
                   const float* __restrict__ kpp,
                   const float* __restrict__ kpw,
                   const float* __restrict__ fsum,
                   float* __restrict__ out,
                   float inv_ext)
{
    __shared__ float s_nx[KNb][136];        // gathered neighbor features (pad 128->136)
    __shared__ float s_infl[16][36];        // influence^T [p][k]        (pad 32->36)
    __shared__ float s_w[16][Pk][132];      // weighted  [pt][p][c]       (pad 128->132)
    __shared__ int   s_idx[KNb];
    __shared__ float s_nnum[16];
    __shared__ int   s_cnt;

    const int tid  = threadIdx.x;
    const int wave = tid >> 5;
    const int lane = tid & 31;
    const int half = lane >> 4;
    const int l16  = lane & 15;
    const int n0 = blockIdx.x * 16;
    const int col0 = wave * 32 + l16;
    const int col1 = col0 + 16;

    for (int j = 0; j < 16; ++j) {
        const int n = n0 + j;
        const float cx = pts[n * 3 + 0], cy = pts[n * 3 + 1], cz = pts[n * 3 + 2];
        if (tid == 0) s_cnt = 0;
        __syncthreads();                      // also fences prior iteration's LDS reads
        if (tid < KNb) {                      // exactly wave 0 -> wave-uniform branch
            const int idx = nbr[n * KNb + tid];
            s_idx[tid] = idx;
            const float dx = pts[idx * 3 + 0] - cx;
            const float dy = pts[idx * 3 + 1] - cy;
            const float dz = pts[idx * 3 + 2] - cz;
            const float d2 = dx * dx + dy * dy + dz * dz;
            for (int p = 0; p < Pk; ++p) {
                const float kx = kpp[p * 3 + 0], ky = kpp[p * 3 + 1], kz = kpp[p * 3 + 2];
                const float sq = d2 + kx * kx + ky * ky + kz * kz
                               - 2.0f * (dx * kx + dy * ky + dz * kz);
                const float dd = sqrtf(fmaxf(sq, 0.0f));
                s_infl[p][tid] = fmaxf(1.0f - dd * inv_ext, 0.0f);
            }
            s_infl[15][tid] = 0.0f;           // zero-pad P 15 -> 16
            if (fsum[idx] > 0.0f) atomicAdd(&s_cnt, 1);
        }
        __syncthreads();
        for (int kk = 0; kk < KNb; ++kk)      // coalesced gather: tid = channel
            s_nx[kk][tid] = feat[(size_t)s_idx[kk] * Cch + tid];
        if (tid == 0) s_nnum[j] = (float)(s_cnt > 1 ? s_cnt : 1);
        __syncthreads();

        // Stage 1: shared A fragment feeds both column tiles
        {
            v8f w0 = {}, w1 = {};
#pragma unroll
            for (int k = 0; k < KNb; k += 4) {
                const int ka = k + 2 * half;
                v2f a, b0, b1;
                a.x  = s_infl[l16][ka];   a.y  = s_infl[l16][ka + 1];
                b0.x = s_nx[ka][col0];    b0.y = s_nx[ka + 1][col0];
                b1.x = s_nx[ka][col1];    b1.y = s_nx[ka + 1][col1];
                w0 = WMMA_F32(a, b0, w0);
                w1 = WMMA_F32(a, b1, w1);
            }
#pragma unroll
            for (int r = 0; r < 8; ++r) {
                const int pr = r + 8 * half;
                if (pr < Pk) {
                    s_w[j][pr][col0] = w0[r];
                    s_w[j][pr][col1] = w1[r];
                }
            }
        }
    }
    __syncthreads();

    // Stage 2: conv[m,d] = sum_p weighted[m,p,:] @ kpw[p]   (M = 16 points)
    {
        v8f acc0 = {}, acc1 = {};
        for (int p = 0; p < Pk; ++p) {
            const float* W = kpw + (size_t)p * Cch * Cch;
            if (p + 1 < Pk)   // pull next weight block toward L2 (global_prefetch_b8)
                __builtin_prefetch(kpw + (size_t)(p + 1) * Cch * Cch + (size_t)tid * 128, 0, 1);
#pragma unroll 2
            for (int k = 0; k < Cch; k += 4) {
                const int ka = k + 2 * half;
                v2f a, b0, b1;
                a.x  = s_w[l16][p][ka];        a.y  = s_w[l16][p][ka + 1];
                b0.x = W[ka * Cch + col0];     b0.y = W[(ka + 1) * Cch + col0];
                b1.x = W[ka * Cch + col1];     b1.y = W[(ka + 1) * Cch + col1];
                acc0 = WMMA_F32(a, b0, acc0);
                acc1 = WMMA_F32(a, b1, acc1);
            }
        }
#pragma unroll
        for (int r = 0; r < 8; ++r) {
            const int m = r + 8 * half;
            const float inv = 1.0f / s_nnum[m];
            out[(size_t)(n0 + m) * Cch + col0] = acc0[r] * inv;
            out[(size_t)(n0 + m) * Cch + col1] = acc1[r] * inv;
        }
    }
}

// ---------------------------------------------------------------------------
// Final linear head (Kd -> 1) + sigmoid; one wave per row, coalesced.
// ---------------------------------------------------------------------------
__global__ __launch_bounds__(128)
void sig_head(const float* __restrict__ X, int Kd,
              const float* __restrict__ w, const float* __restrict__ b,
              float* __restrict__ out, int M)
{
    const int row  = blockIdx.x * 4 + (threadIdx.x >> 5);
    const int lane = threadIdx.x & 31;
    if (row >= M) return;
    float s = 0.0f;
    for (int k = lane; k < Kd; k += 32) s += X[(size_t)row * Kd + k] * w[k];
    for (int off = 16; off; off >>= 1) s += __shfl_down(s, off, 32);
    if (lane == 0) out[row] = 1.0f / (1.0f + expf(-(s + b[0])));
}

// ---------------------------------------------------------------------------
extern "C" void kernel_launch(void* const* d_in, const int* in_sizes, int n_in,
                              void* d_out, int out_size, void* d_ws, size_t ws_size,
                              hipStream_t stream)
{
    (void)in_sizes; (void)n_in; (void)out_size; (void)ws_size;

    const float* points    = (const float*)d_in[0];
    const float* cov       = (const float*)d_in[1];
    const int*   nbr       = (const int*)  d_in[2];
    const float* pre_w     = (const float*)d_in[3];
    const float* pre_bn_g  = (const float*)d_in[5];   // linear biases before BN cancel
    const float* pre_bn_b  = (const float*)d_in[6];
    const float* kpp       = (const float*)d_in[7];
    const float* kpw       = (const float*)d_in[8];
    const float* pos_w1    = (const float*)d_in[9];
    const float* pos_bn1_g = (const float*)d_in[11];
    const float* pos_bn1_b = (const float*)d_in[12];
    const float* pos_w2    = (const float*)d_in[13];
    const float* pos_bn2_g = (const float*)d_in[15];
    const float* pos_bn2_b = (const float*)d_in[16];
    const float* agg_w1    = (const float*)d_in[17];
    const float* agg_bn1_g = (const float*)d_in[19];
    const float* agg_bn1_b = (const float*)d_in[20];
    const float* agg_w2    = (const float*)d_in[21];
    const float* agg_bn2_g = (const float*)d_in[23];
    const float* agg_bn2_b = (const float*)d_in[24];
    const float* keep_w1   = (const float*)d_in[25];
    const float* keep_bn_g = (const float*)d_in[27];
    const float* keep_bn_b = (const float*)d_in[28];
    const float* keep_w2   = (const float*)d_in[29];
    const float* keep_b2   = (const float*)d_in[30];
    const float* uniq_w1   = (const float*)d_in[31];
    const float* uniq_bn_g = (const float*)d_in[33];
    const float* uniq_bn_b = (const float*)d_in[34];
    const float* uniq_w2   = (const float*)d_in[35];
    const float* uniq_b2   = (const float*)d_in[36];

    float* ws    = (float*)d_ws;
    float* feat  = ws;                         // [N,128]
    float* fsum  = feat + (size_t)Npts * 128;  // [N]
    float* kpc   = fsum + Npts;                // [N,128]
    float* rawb  = kpc  + (size_t)Npts * 128;  // [N,256] GEMM output scratch
    float* bufB  = rawb + (size_t)Npts * 256;  // [N,256] activation scratch
    float* bufC  = bufB + (size_t)Npts * 256;  // [N,256] pos embedding
    float* bufD  = bufC + (size_t)Npts * 256;  // [N,256] agg features
    float* stats = bufD + (size_t)Npts * 256;  // [512]
    float* part  = stats + 512;                // [256*2*256] reduction partials

    auto gemm = [&](const float* A, int K, const float* B, int ldb, int Nc,
                    float* C, int acc) {
        dim3 g(Npts / 64, (Nc + 63) / 64);
        gemm_f32_wmma<<<g, 128, 0, stream>>>(A, K, B, ldb, C, Nc, Npts, K, Nc, acc);
    };
    auto bn = [&](const float* raw, float* out, const float* g, const float* b, int Cc) {
        colstats_part<<<256, 256, 0, stream>>>(raw, Npts, Cc, part);
        colstats_final<<<(Cc + 63) / 64, 64, 0, stream>>>(part, Cc, Npts, stats);
        const int total = Npts * Cc;
        bn_lrelu<<<(total + 255) / 256, 256, 0, stream>>>(raw, out, g, b, stats, total, Cc);
    };

    // 1) pre-activation on covariances
    gemm(cov, 9, pre_w, 128, 128, rawb, 0);
    bn(rawb, feat, pre_bn_g, pre_bn_b, 128);
    rowsum_kernel<<<Npts / 4, 128, 0, stream>>>(feat, fsum);

    // 2) KPConv.  inv_ext = 1/KP_EXTENT = (15^(1/3)-1)/1.5
    const float inv_ext = (float)((cbrt(15.0) - 1.0) / 1.5);
    kpconv_kernel<<<Npts / 16, 128, 0, stream>>>(feat, points, nbr, kpp, kpw,
                                                 fsum, kpc, inv_ext);

    // 3) position MLP  3 -> 64 -> 256       (result in bufC)
    gemm(points, 3, pos_w1, 64, 64, rawb, 0);
    bn(rawb, bufB, pos_bn1_g, pos_bn1_b, 64);
    gemm(bufB, 64, pos_w2, 256, 256, rawb, 0);
    bn(rawb, bufC, pos_bn2_g, pos_bn2_b, 256);

    // 4) aggregation MLP 128 -> 256 -> 256  (result in bufD)
    gemm(kpc, 128, agg_w1, 256, 256, rawb, 0);
    bn(rawb, bufB, agg_bn1_g, agg_bn1_b, 256);
    gemm(bufB, 256, agg_w2, 256, 256, rawb, 0);
    bn(rawb, bufD, agg_bn2_g, agg_bn2_b, 256);

    // 5) keep head -> match_score
    gemm(bufD, 256, keep_w1, 64, 64, rawb, 0);
    bn(rawb, bufB, keep_bn_g, keep_bn_b, 64);
    sig_head<<<Npts / 4, 128, 0, stream>>>(bufB, 64, keep_w2, keep_b2,
                                           (float*)d_out, Npts);

    // 6) uniq head: concat(pos, agg) @ uniq_w1 done as two accumulated GEMMs
    gemm(bufC, 256, uniq_w1, 128, 128, rawb, 0);                        // pos @ top half
    gemm(bufD, 256, uniq_w1 + (size_t)256 * 128, 128, 128, rawb, 1);    // agg @ bottom
    bn(rawb, bufB, uniq_bn_g, uniq_bn_b, 128);
    sig_head<<<Npts / 4, 128, 0, stream>>>(bufB, 128, uniq_w2, uniq_b2,
                                           (float*)d_out + Npts, Npts);

    // 7) neighbors_index passthrough (raw int bits into the float output buffer)
    hipMemcpyAsync((char*)d_out + (size_t)2 * Npts * sizeof(float), d_in[2],
                   (size_t)Npts * KNb * sizeof(int), hipMemcpyDeviceToDevice, stream);
}